// TextGraphNN_86861418594784
// MI455X (gfx1250) — compile-verified
//
#include <hip/hip_runtime.h>
#include <hip/hip_bf16.h>
#include <math.h>

// ---------------------------------------------------------------------------
// Problem constants (from reference)
// ---------------------------------------------------------------------------
#define N_NODES   100000
#define N_EDGES   1600000
#define HDIM      128
#define NWIDTH    512
#define NGRAPH    256
#define NLAYER    3
#define NEG_SLOPE 0.2f
#define BN_EPS    1e-5f

typedef __attribute__((ext_vector_type(2))) float v2f;
typedef __attribute__((ext_vector_type(8))) float v8f;

// Use CDNA5 async global->LDS copies when the toolchain declares them.
#ifndef __has_builtin
#define __has_builtin(x) 0
#endif
#if __has_builtin(__builtin_amdgcn_global_load_async_to_lds_b128) && \
    __has_builtin(__builtin_amdgcn_s_wait_asynccnt)
#define USE_ASYNC_LDS 1
#else
#define USE_ASYNC_LDS 0
#endif

#if USE_ASYNC_LDS
// Builtin prototype (from clang diagnostics): pointers to 16-byte int vectors
// in AS1 (global) and AS3 (LDS), plus two immediate ints (offset, cpol).
typedef int v4i __attribute__((vector_size(16)));
typedef v4i __attribute__((address_space(1)))* as1_v4i_p;
typedef v4i __attribute__((address_space(3)))* as3_v4i_p;
#endif

// Copy 16 contiguous bytes global -> LDS.
__device__ __forceinline__ void copy16_g2l(const float* g, float* l) {
#if USE_ASYNC_LDS
  __builtin_amdgcn_global_load_async_to_lds_b128(
      (as1_v4i_p)const_cast<float*>(g), (as3_v4i_p)l,
      /*offset=*/0, /*cpol=*/0);
#else
  *(float4*)l = *(const float4*)g;
#endif
}

__device__ __forceinline__ void lds_copy_fence() {
#if USE_ASYNC_LDS
  __builtin_amdgcn_s_wait_asynccnt(0);
#endif
  __syncthreads();
}

// Monotonic float<->uint mapping so we can do float segment-max with atomicMax.
__device__ __forceinline__ unsigned f2ord(float f) {
  unsigned u = __float_as_uint(f);
  return (u & 0x80000000u) ? ~u : (u | 0x80000000u);
}
__device__ __forceinline__ float ord2f(unsigned u) {
  return __uint_as_float((u & 0x80000000u) ? (u ^ 0x80000000u) : ~u);
}

// ---------------------------------------------------------------------------
// Fp32 WMMA GEMM: C[M x Nc] = A[M x K] @ B[K x Nc] (+bias, optional ReLU)
// V_WMMA_F32_16X16X4_F32; block = 256 threads = 8 waves covering a 32 x 128
// output tile. Each wave owns two vertically stacked 16x16 tiles sharing the
// same B fragment (halves B traffic per FLOP). The 32 x K A tile is staged in
// LDS via async b128 copies; row stride K+4 floats => 16B-aligned rows and
// conflict-free (stride mod 64 banks = 4) WMMA-side reads.
// Requires M % 32 == 0, K % 16 == 0, Nc == 128.
// ---------------------------------------------------------------------------
__global__ __launch_bounds__(256) void gemm_wmma_f32(
    const float* __restrict__ A, const float* __restrict__ B,
    float* __restrict__ C, const float* __restrict__ bias,
    int M, int K, int Nc, int relu)
{
  __shared__ float sA[32 * (NWIDTH + 4)];
  const int tid  = threadIdx.x;
  const int m0   = blockIdx.x * 32;
  const int ldsK = K + 4;
  const int K4   = K >> 2;

  // Stage 32 x K A tile into LDS, 16B per thread per step.
  for (int i = tid; i < 32 * K4; i += 256) {
    int r = i / K4;
    int c = (i - r * K4) << 2;
    copy16_g2l(A + (size_t)(m0 + r) * K + c, sA + r * ldsK + c);
  }
  lds_copy_fence();

  const int wave  = tid >> 5;
  const int lane  = tid & 31;
  const int nl    = lane & 15;   // A: M index within tile; B: N index
  const int khalf = lane >> 4;   // selects K pair {0,1} vs {2,3}
  const int n0    = wave * 16;

  v8f acc0 = {}, acc1 = {};
  #pragma unroll 4
  for (int k = 0; k < K; k += 4) {
    const int ka = k + 2 * khalf;
    v2f a0, a1, b;
    a0.x = sA[nl * ldsK + ka];
    a0.y = sA[nl * ldsK + ka + 1];
    a1.x = sA[(nl + 16) * ldsK + ka];
    a1.y = sA[(nl + 16) * ldsK + ka + 1];
    b.x  = B[(size_t)ka * Nc + n0 + nl];
    b.y  = B[(size_t)(ka + 1) * Nc + n0 + nl];
    acc0 = __builtin_amdgcn_wmma_f32_16x16x4_f32(
        false, a0, false, b, (short)0, acc0, false, false);
    acc1 = __builtin_amdgcn_wmma_f32_16x16x4_f32(
        false, a1, false, b, (short)0, acc1, false, false);
  }

  // C/D layout: VGPR v holds M = v + 8*khalf, N = lane & 15.
  const int ccol  = n0 + nl;
  const float bv  = bias ? bias[ccol] : 0.0f;
  const int rb0   = m0 + khalf * 8;
  #pragma unroll
  for (int v = 0; v < 8; ++v) {
    float x0 = acc0[v] + bv;
    float x1 = acc1[v] + bv;
    if (relu) { x0 = fmaxf(x0, 0.0f); x1 = fmaxf(x1, 0.0f); }
    C[(size_t)(rb0 + v) * Nc + ccol]      = x0;
    C[(size_t)(rb0 + 16 + v) * Nc + ccol] = x1;
  }
}

// ---------------------------------------------------------------------------
// Per-node dot products with the two attention vectors (wave per node).
// ---------------------------------------------------------------------------
__global__ __launch_bounds__(256) void node_dots_k(
    const float* __restrict__ h, const float* __restrict__ as,
    const float* __restrict__ ad, float* __restrict__ osrc,
    float* __restrict__ odst, int n)
{
  int node = blockIdx.x * 8 + (threadIdx.x >> 5);
  if (node >= n) return;
  int lane = threadIdx.x & 31;
  float s1 = 0.f, s2 = 0.f;
  #pragma unroll
  for (int f = lane; f < HDIM; f += 32) {
    float hv = h[(size_t)node * HDIM + f];
    s1 += hv * as[f];
    s2 += hv * ad[f];
  }
  #pragma unroll
  for (int off = 16; off > 0; off >>= 1) {
    s1 += __shfl_xor(s1, off, 32);
    s2 += __shfl_xor(s2, off, 32);
  }
  if (lane == 0) { osrc[node] = s1; odst[node] = s2; }
}

// Per-node dot with a single vector + scalar bias (gate scalar).
__global__ __launch_bounds__(256) void rowdot_k(
    const float* __restrict__ mat, const float* __restrict__ vec,
    const float* __restrict__ bscal, float* __restrict__ outv, int n)
{
  int node = blockIdx.x * 8 + (threadIdx.x >> 5);
  if (node >= n) return;
  int lane = threadIdx.x & 31;
  float s = 0.f;
  #pragma unroll
  for (int f = lane; f < HDIM; f += 32)
    s += mat[(size_t)node * HDIM + f] * vec[f];
  #pragma unroll
  for (int off = 16; off > 0; off >>= 1) s += __shfl_xor(s, off, 32);
  if (lane == 0) outv[node] = s + bscal[0];
}

// ---------------------------------------------------------------------------
// Edge-softmax passes. Edge e < E uses edge_index; e >= E is self loop e-E.
// ---------------------------------------------------------------------------
__device__ __forceinline__ void edge_sd(const int* ei, int e, int& s, int& d) {
  if (e < N_EDGES) { s = ei[e]; d = ei[N_EDGES + e]; }
  else             { s = e - N_EDGES; d = s; }
}

__global__ __launch_bounds__(256) void edge_max_k(
    const int* __restrict__ ei, const float* __restrict__ asrc,
    const float* __restrict__ adst, unsigned* __restrict__ mbuf, int etot)
{
  int e = blockIdx.x * blockDim.x + threadIdx.x;
  if (e >= etot) return;
  int s, d; edge_sd(ei, e, s, d);
  float v = asrc[s] + adst[d];
  v = (v > 0.f) ? v : NEG_SLOPE * v;
  atomicMax(&mbuf[d], f2ord(v));
}

__global__ __launch_bounds__(256) void edge_exp_k(
    const int* __restrict__ ei, const float* __restrict__ asrc,
    const float* __restrict__ adst, const unsigned* __restrict__ mbuf,
    float* __restrict__ den, float* __restrict__ exq, int etot)
{
  int e = blockIdx.x * blockDim.x + threadIdx.x;
  if (e >= etot) return;
  int s, d; edge_sd(ei, e, s, d);
  float v = asrc[s] + adst[d];
  v = (v > 0.f) ? v : NEG_SLOPE * v;
  float ex = __expf(v - ord2f(mbuf[d]));
  exq[e] = ex;
  atomicAdd(&den[d], ex);
}

// Weighted scatter: out[dst] += h[src] * coef. One wave per edge; each lane
// handles 4 contiguous floats (one global_load_b128 per lane => one full
// 512B row per wave), then 4 float atomic adds into the destination row.
__global__ __launch_bounds__(256) void edge_agg_k(
    const int* __restrict__ ei, const float* __restrict__ h,
    const float* __restrict__ exq, const float* __restrict__ den,
    float* __restrict__ outb, int etot)
{
  int e = blockIdx.x * 8 + (threadIdx.x >> 5);
  if (e >= etot) return;
  int lane = threadIdx.x & 31;
  int s, d; edge_sd(ei, e, s, d);
  float coef = exq[e] / den[d];
  int f0 = lane << 2;
  const float4 hv = *(const float4*)(h + (size_t)s * HDIM + f0);
  float* od = outb + (size_t)d * HDIM + f0;
  atomicAdd(&od[0], hv.x * coef);
  atomicAdd(&od[1], hv.y * coef);
  atomicAdd(&od[2], hv.z * coef);
  atomicAdd(&od[3], hv.w * coef);
}

// ---------------------------------------------------------------------------
// BatchNorm over N rows, 128 columns. Stats via per-block partial sums
// (coalesced: 128 threads sweep rows), then elementwise apply + ReLU in-place.
// Note: the reference adds `bias` before BN; a per-column constant shift
// cancels exactly in (x - mean), so bias is skipped.
// ---------------------------------------------------------------------------
__global__ __launch_bounds__(128) void bn_stats_k(
    const float* __restrict__ x, float* __restrict__ colsum,
    float* __restrict__ colss, int n)
{
  int col = threadIdx.x;
  int r0 = blockIdx.x * 256;
  int r1 = min(r0 + 256, n);
  float s = 0.f, ss = 0.f;
  for (int r = r0; r < r1; ++r) {
    float v = x[(size_t)r * HDIM + col];
    s += v; ss += v * v;
  }
  atomicAdd(&colsum[col], s);
  atomicAdd(&colss[col], ss);
}

__global__ __launch_bounds__(256) void bn_apply_k(
    float* __restrict__ x, const float* __restrict__ colsum,
    const float* __restrict__ colss, const float* __restrict__ gamma,
    const float* __restrict__ beta, int n)
{
  size_t idx = (size_t)blockIdx.x * blockDim.x + threadIdx.x;
  if (idx >= (size_t)n * HDIM) return;
  int col = (int)(idx & (HDIM - 1));
  float inv_n = 1.0f / (float)n;
  float mu  = colsum[col] * inv_n;
  float var = colss[col] * inv_n - mu * mu;
  float v = (x[idx] - mu) * rsqrtf(var + BN_EPS) * gamma[col] + beta[col];
  x[idx] = fmaxf(v, 0.f);
}

// ---------------------------------------------------------------------------
// Graph-level softmax pooling.
// ---------------------------------------------------------------------------
__global__ __launch_bounds__(256) void pool_max_k(
    const float* __restrict__ g, const int* __restrict__ batch,
    unsigned* __restrict__ gm, int n)
{
  int i = blockIdx.x * blockDim.x + threadIdx.x;
  if (i >= n) return;
  atomicMax(&gm[batch[i]], f2ord(g[i]));
}

__global__ __launch_bounds__(256) void pool_expsum_k(
    const float* __restrict__ g, const int* __restrict__ batch,
    const unsigned* __restrict__ gm, float* __restrict__ gex,
    float* __restrict__ gden, int n)
{
  int i = blockIdx.x * blockDim.x + threadIdx.x;
  if (i >= n) return;
  int b = batch[i];
  float e = __expf(g[i] - ord2f(gm[b]));
  gex[i] = e;
  atomicAdd(&gden[b], e);
}

__global__ __launch_bounds__(256) void pool_scatter_k(
    const float* __restrict__ x, const int* __restrict__ batch,
    const float* __restrict__ gex, const float* __restrict__ gden,
    float* __restrict__ pooled, int n)
{
  int node = blockIdx.x * 8 + (threadIdx.x >> 5);
  if (node >= n) return;
  int lane = threadIdx.x & 31;
  int b = batch[node];
  float dv = gden[b];
  if (dv == 0.f) dv = 1.f;
  float coef = gex[node] / dv;
  int f0 = lane << 2;
  const float4 xv = *(const float4*)(x + (size_t)node * HDIM + f0);
  float* pr = pooled + (size_t)b * HDIM + f0;
  atomicAdd(&pr[0], xv.x * coef);
  atomicAdd(&pr[1], xv.y * coef);
  atomicAdd(&pr[2], xv.z * coef);
  atomicAdd(&pr[3], xv.w * coef);
}

// logits = pooled @ cls_w + cls_b; out = log_softmax(logits). Wave per graph.
__global__ __launch_bounds__(256) void classify_k(
    const float* __restrict__ pooled, const float* __restrict__ cw,
    const float* __restrict__ cb, float* __restrict__ out, int g)
{
  int gi = blockIdx.x * 8 + (threadIdx.x >> 5);
  if (gi >= g) return;
  int lane = threadIdx.x & 31;
  float l0 = 0.f, l1 = 0.f;
  #pragma unroll
  for (int f = lane; f < HDIM; f += 32) {
    float p = pooled[(size_t)gi * HDIM + f];
    l0 += p * cw[2 * f];
    l1 += p * cw[2 * f + 1];
  }
  #pragma unroll
  for (int off = 16; off > 0; off >>= 1) {
    l0 += __shfl_xor(l0, off, 32);
    l1 += __shfl_xor(l1, off, 32);
  }
  if (lane == 0) {
    l0 += cb[0]; l1 += cb[1];
    float mx = fmaxf(l0, l1);
    float lse = mx + logf(expf(l0 - mx) + expf(l1 - mx));
    out[2 * gi]     = l0 - lse;
    out[2 * gi + 1] = l1 - lse;
  }
}

// ---------------------------------------------------------------------------
// Host-side orchestration.
// ---------------------------------------------------------------------------
extern "C" void kernel_launch(void* const* d_in, const int* in_sizes, int n_in,
                              void* d_out, int out_size, void* d_ws, size_t ws_size,
                              hipStream_t stream) {
  (void)in_sizes; (void)n_in; (void)out_size; (void)ws_size;
  const float* x        = (const float*)d_in[0];   // N x 512
  const int*   ei       = (const int*)  d_in[1];   // 2 x E
  // d_in[2] = edge_attr (unused: reference discards edge_emb lookup)
  const int*   batch    = (const int*)  d_in[3];   // N
  // d_in[4] = edge_emb (unused)
  const float* W0       = (const float*)d_in[5];   // 512 x 128
  const float* W_rest   = (const float*)d_in[6];   // 2 x 128 x 128
  const float* att_src  = (const float*)d_in[7];   // 3 x 128
  const float* att_dst  = (const float*)d_in[8];   // 3 x 128
  // d_in[9] = bias (cancels in BatchNorm; unused)
  const float* gamma    = (const float*)d_in[10];  // 3 x 128
  const float* beta     = (const float*)d_in[11];  // 3 x 128
  const float* gate_w1  = (const float*)d_in[12];  // 128 x 128
  const float* gate_b1  = (const float*)d_in[13];  // 128
  const float* gate_w2  = (const float*)d_in[14];  // 128
  const float* gate_b2  = (const float*)d_in[15];  // 1
  const float* cls_w    = (const float*)d_in[16];  // 128 x 2
  const float* cls_b    = (const float*)d_in[17];  // 2
  float* out = (float*)d_out;                      // 256 x 2

  const int n = N_NODES, etot = N_EDGES + N_NODES;

  // Workspace carve-up (floats).
  float* base   = (float*)d_ws;
  float* hbuf   = base;                               // N*H  (GEMM out / gate hidden)
  float* abuf   = hbuf + (size_t)n * HDIM;            // N*H  (aggregated / layer out)
  float* asrc   = abuf + (size_t)n * HDIM;            // N
  float* adst   = asrc + n;                           // N
  unsigned* mbuf = (unsigned*)(adst + n);             // N
  float* den    = (float*)mbuf + n;                   // N
  float* exq    = den + n;                            // E+N
  float* colsum = exq + etot;                         // H
  float* colss  = colsum + HDIM;                      // H
  float* gscal  = colss + HDIM;                       // N
  float* gex    = gscal + n;                          // N
  unsigned* gm  = (unsigned*)(gex + n);               // G
  float* gden   = (float*)gm + NGRAPH;                // G
  float* pooled = gden + NGRAPH;                      // G*H

  const int eblk = (etot + 255) / 256;
  const int nblk = (n + 255) / 256;

  const float* cur = x;
  int K = NWIDTH;
  for (int l = 0; l < NLAYER; ++l) {
    const float* W = (l == 0) ? W0 : (W_rest + (size_t)(l - 1) * HDIM * HDIM);
    // h = cur @ W  (fp32 WMMA, 32-row blocks)
    gemm_wmma_f32<<<n / 32, 256, 0, stream>>>(cur, W, hbuf, nullptr, n, K, HDIM, 0);
    // attention scalars
    node_dots_k<<<(n + 7) / 8, 256, 0, stream>>>(hbuf, att_src + l * HDIM,
                                                 att_dst + l * HDIM, asrc, adst, n);
    // edge softmax + weighted aggregation
    (void)hipMemsetAsync(mbuf, 0, (size_t)n * 4, stream);
    (void)hipMemsetAsync(den,  0, (size_t)n * 4, stream);
    (void)hipMemsetAsync(abuf, 0, (size_t)n * HDIM * 4, stream);
    edge_max_k<<<eblk, 256, 0, stream>>>(ei, asrc, adst, mbuf, etot);
    edge_exp_k<<<eblk, 256, 0, stream>>>(ei, asrc, adst, mbuf, den, exq, etot);
    edge_agg_k<<<(etot + 7) / 8, 256, 0, stream>>>(ei, hbuf, exq, den, abuf, etot);
    // BatchNorm + ReLU (in place on abuf)
    (void)hipMemsetAsync(colsum, 0, HDIM * 4, stream);
    (void)hipMemsetAsync(colss,  0, HDIM * 4, stream);
    bn_stats_k<<<(n + 255) / 256, 128, 0, stream>>>(abuf, colsum, colss, n);
    bn_apply_k<<<(int)(((size_t)n * HDIM + 255) / 256), 256, 0, stream>>>(
        abuf, colsum, colss, gamma + l * HDIM, beta + l * HDIM, n);
    cur = abuf;
    K = HDIM;
  }

  // Gate MLP: hbuf = relu(abuf @ gate_w1 + gate_b1); gscal = hbuf @ gate_w2 + gate_b2
  gemm_wmma_f32<<<n / 32, 256, 0, stream>>>(abuf, gate_w1, hbuf, gate_b1, n, HDIM, HDIM, 1);
  rowdot_k<<<(n + 7) / 8, 256, 0, stream>>>(hbuf, gate_w2, gate_b2, gscal, n);

  // Per-graph softmax pooling.
  (void)hipMemsetAsync(gm,     0, NGRAPH * 4, stream);
  (void)hipMemsetAsync(gden,   0, NGRAPH * 4, stream);
  (void)hipMemsetAsync(pooled, 0, (size_t)NGRAPH * HDIM * 4, stream);
  pool_max_k<<<nblk, 256, 0, stream>>>(gscal, batch, gm, n);
  pool_expsum_k<<<nblk, 256, 0, stream>>>(gscal, batch, gm, gex, gden, n);
  pool_scatter_k<<<(n + 7) / 8, 256, 0, stream>>>(abuf, batch, gex, gden, pooled, n);

  // Classifier + log-softmax.
  classify_k<<<NGRAPH / 8, 256, 0, stream>>>(pooled, cls_w, cls_b, out, NGRAPH);
}